// CCSOFT_17763984736782
// MI455X (gfx1250) — compile-verified
//
#include <hip/hip_runtime.h>
#include <hip/hip_bf16.h>

// Problem constants from the reference
#define LQ 64
#define BQ 32
#define MQ 127          // 2*L-1
#define M3 ((size_t)MQ * MQ * MQ)          // 2,048,383
#define NPLANE ((size_t)BQ * M3)           // 65,548,256 floats per re/im plane
#define NGROUPS 32258                      // (32*127*127)/16 row groups per pass
#define PERSIST_BLOCKS 4096

typedef float v2f __attribute__((ext_vector_type(2)));
typedef float v8f __attribute__((ext_vector_type(8)));
typedef __attribute__((address_space(3))) float lds_float;

// ---------------------------------------------------------------------------
// Kernel 0: build the 127-pt inverse-DFT matrix, zero-padded to 128x128.
// D[j,i] = exp(+2*pi*i*j/127)/127  (symmetric; row/col 127 = 0 pad)
// ---------------------------------------------------------------------------
__global__ __launch_bounds__(256) void CCSOFT_twiddle_kernel(
    float* __restrict__ dt_re, float* __restrict__ dt_im) {
  int idx = blockIdx.x * 256 + threadIdx.x;           // 0 .. 16383
  if (idx >= 128 * 128) return;
  int j = idx >> 7, i = idx & 127;
  float vr = 0.f, vi = 0.f;
  if (i < MQ && j < MQ) {
    int ph = (i * j) % MQ;                            // exact modular reduction
    float ang = 6.283185307179586f * (float)ph / 127.f;
    float s, c;
    __sincosf(ang, &s, &c);
    vr = c * (1.0f / 127.0f);
    vi = s * (1.0f / 127.0f);
  }
  dt_re[idx] = vr;
  dt_im[idx] = vi;
}

// ---------------------------------------------------------------------------
// Kernel 1: banded Wigner contraction.
// cc[b,m,k,n] = sum_l wig[l,m,k,n] * f[b,l,m] * conj(g[b,l,n])
// One block per (m,k); 128 threads over n; 32 complex accumulators per thread.
// Band structure: wig row is zero unless l >= max(|m-63|,|k-63|); additionally
// lanes of a wave share a minimum |n-63| so each wave can raise its own l0.
// ---------------------------------------------------------------------------
__global__ __launch_bounds__(128) void CCSOFT_contract_kernel(
    const float* __restrict__ f_re, const float* __restrict__ f_im,
    const float* __restrict__ g_re, const float* __restrict__ g_im,
    const float* __restrict__ wig,
    float* __restrict__ cc_re, float* __restrict__ cc_im) {
  __shared__ __align__(16) float fsh_re[LQ * BQ];
  __shared__ __align__(16) float fsh_im[LQ * BQ];
  const int m = blockIdx.x, k = blockIdx.y;
  const int t = threadIdx.x;

  // stage f[:, :, m] (broadcast operand) into LDS, layout [l][b]
  for (int idx = t; idx < LQ * BQ; idx += 128) {
    int l = idx >> 5, b = idx & 31;
    fsh_re[idx] = f_re[(b * LQ + l) * MQ + m];
    fsh_im[idx] = f_im[(b * LQ + l) * MQ + m];
  }
  __syncthreads();

  const int n = (t < MQ) ? t : (MQ - 1);
  float acc_re[BQ], acc_im[BQ];
#pragma unroll
  for (int b = 0; b < BQ; ++b) { acc_re[b] = 0.f; acc_im[b] = 0.f; }

  int am = m - 63; am = am < 0 ? -am : am;
  int ak = k - 63; ak = ak < 0 ? -ak : ak;
  int l0 = am > ak ? am : ak;
  const int wv = t >> 5;                               // wave id: n in [32w,32w+31]
  const int wvlmin = (wv == 0) ? 32 : (wv == 1 ? 0 : (wv == 2 ? 1 : 33));
  if (wvlmin > l0) l0 = wvlmin;

  const size_t wigbase = ((size_t)m * MQ + k) * MQ + n;
  for (int l = l0; l < LQ; ++l) {
    float w = wig[(size_t)l * M3 + wigbase];           // coalesced over n
    if (l + 1 < LQ)
      __builtin_prefetch(wig + (size_t)(l + 1) * M3 + wigbase, 0, 1);
#pragma unroll 8
    for (int b = 0; b < BQ; ++b) {
      float fr = fsh_re[l * BQ + b], fi = fsh_im[l * BQ + b];   // LDS broadcast
      float gr = g_re[(b * LQ + l) * MQ + n];                   // L2-resident
      float gi = g_im[(b * LQ + l) * MQ + n];
      acc_re[b] = fmaf(w, fmaf(fr, gr, fi * gi), acc_re[b]);
      acc_im[b] = fmaf(w, fmaf(fi, gr, -fr * gi), acc_im[b]);
    }
  }

  if (t < MQ) {
#pragma unroll 4
    for (int b = 0; b < BQ; ++b) {
      size_t o = (((size_t)b * MQ + m) * MQ + k) * MQ + n;      // [b,m,k,n]
      cc_re[o] = acc_re[b];
      cc_im[o] = acc_im[b];
    }
  }
}

// ---------------------------------------------------------------------------
// Async staging: stream 16 input rows straight into LDS (no VGPR staging),
// tracked on ASYNCcnt. Pad columns (>=127) are pre-zeroed once and never
// written by these loads.
// ---------------------------------------------------------------------------
#define SPITCH 132   // LDS row pitch (floats): conflict-free for strided reads

__device__ __forceinline__ void CCSOFT_async_stage(
    const float* __restrict__ src, float* lds_dst, long r0, int t) {
  for (int idx = t; idx < 16 * SPITCH; idx += 256) {
    int row = idx / SPITCH, col = idx - row * SPITCH;
    if (col < MQ) {
      const float* gp = src + (r0 + row) * MQ + col;
      unsigned lo = (unsigned)(unsigned long long)(lds_float*)(lds_dst + idx);
      asm volatile("global_load_async_to_lds_b32 %0, %1, off"
                   :: "v"(lo), "v"(gp) : "memory");
    }
  }
}

// ---------------------------------------------------------------------------
// Kernel 2: one inverse-DFT pass along the contiguous last axis via WMMA f32.
// Input  layout: [B, D1, D2, 127] (rows r = (b,d1,d2), axis contiguous)
// Output layout: [B, 127, D1, D2] (axis rotated to front of spatial dims)
// out[j, r] = sum_i D[j,i] * in[r, i]   (complex x complex, D symmetric)
// Persistent blocks: per-lane Dt slice preloaded into registers once;
// double-buffered LDS tiles filled by async global->LDS loads overlap the
// 128-WMMA compute of the previous row group.
// ---------------------------------------------------------------------------
__global__ __launch_bounds__(256) void CCSOFT_dft_pass_kernel(
    const float* __restrict__ in_re, const float* __restrict__ in_im,
    const float* __restrict__ dt_re, const float* __restrict__ dt_im,
    float* __restrict__ out_re, float* __restrict__ out_im,
    int interleave) {
  __shared__ __align__(16) float sh_re[2][16 * SPITCH];
  __shared__ __align__(16) float sh_im[2][16 * SPITCH];
  const int t = threadIdx.x;
  const int wave = t >> 5, lane = t & 31;
  const int half = lane >> 4, l16 = lane & 15;
  const int j0 = wave * 16;                 // 8 waves cover j tiles 0..127
  const int koff = 2 * half;                // A/B fragment K selector per half

  // zero the pad columns of both buffers once (never touched by async loads)
  for (int idx = t; idx < 16 * SPITCH; idx += 256) {
    int col = idx - (idx / SPITCH) * SPITCH;
    if (col >= MQ) {
      sh_re[0][idx] = 0.f; sh_re[1][idx] = 0.f;
      sh_im[0][idx] = 0.f; sh_im[1][idx] = 0.f;
    }
  }

  // Per-lane Dt fragments are loop-invariant: preload 32 K-steps into VGPRs.
  v2f ar[32], ai[32];
  {
    const float* dtr = dt_re + (j0 + l16) * 128 + koff;
    const float* dti = dt_im + (j0 + l16) * 128 + koff;
#pragma unroll
    for (int s = 0; s < 32; ++s) {
      ar[s] = *(const v2f*)(dtr + 4 * s);
      ai[s] = *(const v2f*)(dti + 4 * s);
    }
  }

  long g = blockIdx.x;
  CCSOFT_async_stage(in_re, &sh_re[0][0], g * 16, t);
  CCSOFT_async_stage(in_im, &sh_im[0][0], g * 16, t);
  int cur = 0;

  for (; g < NGROUPS; g += gridDim.x) {
    asm volatile("s_wait_asynccnt 0x0" ::: "memory");  // own async done
    __syncthreads();                                   // everyone's async done

    long gn = g + gridDim.x;                           // prefetch next group
    if (gn < NGROUPS) {
      CCSOFT_async_stage(in_re, &sh_re[cur ^ 1][0], gn * 16, t);
      CCSOFT_async_stage(in_im, &sh_im[cur ^ 1][0], gn * 16, t);
    }

    const float* shr = &sh_re[cur][0] + l16 * SPITCH + koff;
    const float* shi = &sh_im[cur][0] + l16 * SPITCH + koff;
    v8f rr = {}, ii = {}, ri = {}, ir = {};
#pragma unroll
    for (int s = 0; s < 32; ++s) {
      v2f b_r = *(const v2f*)(shr + 4 * s);            // ds_load_b64
      v2f b_i = *(const v2f*)(shi + 4 * s);
      rr = __builtin_amdgcn_wmma_f32_16x16x4_f32(false, ar[s], false, b_r,
                                                 (short)0, rr, false, false);
      ii = __builtin_amdgcn_wmma_f32_16x16x4_f32(false, ai[s], false, b_i,
                                                 (short)0, ii, false, false);
      ri = __builtin_amdgcn_wmma_f32_16x16x4_f32(false, ar[s], false, b_i,
                                                 (short)0, ri, false, false);
      ir = __builtin_amdgcn_wmma_f32_16x16x4_f32(false, ai[s], false, b_r,
                                                 (short)0, ir, false, false);
    }
    v8f cre = rr - ii;            // (Ar+iAi)(Br+iBi): re = ArBr - AiBi
    v8f cim = ri + ir;            //                   im = ArBi + AiBr

    // C layout: VGPR v -> row j = j0 + v + 8*half, col r = r0 + (lane&15)
    long r = g * 16 + l16;
    long b = r / (MQ * MQ);
    long rem = r - b * (MQ * MQ);
    long d1 = rem / MQ;
    long d2 = rem - d1 * MQ;
#pragma unroll
    for (int v = 0; v < 8; ++v) {
      int j = j0 + v + 8 * half;
      if (j < MQ) {
        long o = ((b * MQ + j) * MQ + d1) * MQ + d2;   // [b, j, d1, d2]
        if (interleave) {                               // final pass: complex64
          float2 val;
          val.x = cre[v];
          val.y = cim[v];
          ((float2*)out_re)[o] = val;                   // coalesced b64 stores
        } else {
          out_re[o] = cre[v];
          out_im[o] = cim[v];
        }
      }
    }
    cur ^= 1;
  }
}

// ---------------------------------------------------------------------------
// Host-side launch: twiddle -> contraction -> 3 WMMA DFT passes.
// Axis rotation per pass returns the layout to [b,m,k,n] after pass 3.
// Workspace: 128KB twiddles + two complex plane-pair buffers (~1.05 GB).
// ---------------------------------------------------------------------------
extern "C" void kernel_launch(void* const* d_in, const int* in_sizes, int n_in,
                              void* d_out, int out_size, void* d_ws, size_t ws_size,
                              hipStream_t stream) {
  const float* f_re = (const float*)d_in[0];
  const float* f_im = (const float*)d_in[1];
  const float* g_re = (const float*)d_in[2];
  const float* g_im = (const float*)d_in[3];
  const float* wig  = (const float*)d_in[4];
  float* out = (float*)d_out;

  float* ws = (float*)d_ws;
  float* dt_re = ws;                       // 128*128
  float* dt_im = ws + 16384;
  float* A_re = ws + 32768;
  float* A_im = A_re + NPLANE;
  float* B_re = A_im + NPLANE;
  float* B_im = B_re + NPLANE;

  // 0) inverse-DFT matrix (zero-padded 128x128)
  CCSOFT_twiddle_kernel<<<64, 256, 0, stream>>>(dt_re, dt_im);

  // 1) banded Wigner contraction -> cc planes in [b,m,k,n]
  dim3 cgrid(MQ, MQ);
  CCSOFT_contract_kernel<<<cgrid, 128, 0, stream>>>(f_re, f_im, g_re, g_im,
                                                    wig, A_re, A_im);

  // 2) three inverse-DFT passes (n, then k, then m), rotating layout each time
  CCSOFT_dft_pass_kernel<<<PERSIST_BLOCKS, 256, 0, stream>>>(
      A_re, A_im, dt_re, dt_im, B_re, B_im, 0);
  CCSOFT_dft_pass_kernel<<<PERSIST_BLOCKS, 256, 0, stream>>>(
      B_re, B_im, dt_re, dt_im, A_re, A_im, 0);
  CCSOFT_dft_pass_kernel<<<PERSIST_BLOCKS, 256, 0, stream>>>(
      A_re, A_im, dt_re, dt_im, out, nullptr, 1);
}